// SetAbstraction_52596169507543
// MI455X (gfx1250) — compile-verified
//
#include <hip/hip_runtime.h>

// ---------------------------------------------------------------------------
// PointNet++ SetAbstraction for MI455X (gfx1250, wave32, WMMA f32 16x16x4)
// ---------------------------------------------------------------------------

typedef float v2f __attribute__((ext_vector_type(2)));
typedef float v8f __attribute__((ext_vector_type(8)));

#define NPTS      4096
#define NBATCH    16
#define NPOINT    1024
#define NSAMPLE   32
#define NQ        (NBATCH * NPOINT)          // 16384 query points
#define MROWS     ((size_t)NQ * NSAMPLE)     // 524288 GEMM rows
#define LDX1      68                          // 67 channels + 1 zero pad (mult of 4)
#define NEWXYZ_N  (NQ * 3)                    // 49152 floats

// ---------------------------------------------------------------------------
// K0: zero a float buffer (graph-capture-safe replacement for memset)
// ---------------------------------------------------------------------------
__global__ void zero_kernel(float* __restrict__ p, int n) {
    int i = blockIdx.x * blockDim.x + threadIdx.x;
    if (i < n) p[i] = 0.f;
}

// ---------------------------------------------------------------------------
// K1: farthest point sampling. One block per batch. Cloud in LDS, distances
// in registers (8 per thread), shuffle argmax + small LDS reduction per step.
// ---------------------------------------------------------------------------
__global__ __launch_bounds__(512) void fps_kernel(const float* __restrict__ xyz,
                                                  float* __restrict__ newxyz,
                                                  int* __restrict__ fps_idx) {
    __shared__ float xs[NPTS], ys[NPTS], zs[NPTS];
    __shared__ float rv[16];
    __shared__ int   ri[16];
    __shared__ float bc[3];
    __shared__ int   bfar;

    const int b   = blockIdx.x;
    const int tid = threadIdx.x;
    const float* base = xyz + (size_t)b * NPTS * 3;

    for (int i = tid; i < NPTS; i += 512) {
        xs[i] = base[i * 3 + 0];
        ys[i] = base[i * 3 + 1];
        zs[i] = base[i * 3 + 2];
    }
    float d[8];
#pragma unroll
    for (int j = 0; j < 8; ++j) d[j] = 1e10f;
    __syncthreads();
    if (tid == 0) { bc[0] = xs[0]; bc[1] = ys[0]; bc[2] = zs[0]; bfar = 0; }
    __syncthreads();

    for (int t = 0; t < NPOINT; ++t) {
        const float cx = bc[0], cy = bc[1], cz = bc[2];
        if (tid == 0) {
            int far = bfar;
            fps_idx[b * NPOINT + t] = far;
            size_t o = (size_t)(b * NPOINT + t) * 3;
            newxyz[o + 0] = cx; newxyz[o + 1] = cy; newxyz[o + 2] = cz;
        }
        float bv = -1.f; int bi = 0;
#pragma unroll
        for (int j = 0; j < 8; ++j) {
            int i = tid + j * 512;
            float dx = xs[i] - cx, dy = ys[i] - cy, dz = zs[i] - cz;
            float dd = dx * dx + dy * dy + dz * dz;
            d[j] = fminf(d[j], dd);
            if (d[j] > bv) { bv = d[j]; bi = i; }
        }
        // wave32 shuffle argmax (ties -> smaller index, matching jnp.argmax)
        for (int off = 16; off > 0; off >>= 1) {
            float v2 = __shfl_down(bv, off);
            int   i2 = __shfl_down(bi, off);
            if (v2 > bv || (v2 == bv && i2 < bi)) { bv = v2; bi = i2; }
        }
        if ((tid & 31) == 0) { rv[tid >> 5] = bv; ri[tid >> 5] = bi; }
        __syncthreads();
        if (tid == 0) {
            float mv = rv[0]; int mi = ri[0];
            for (int k = 1; k < 16; ++k)
                if (rv[k] > mv || (rv[k] == mv && ri[k] < mi)) { mv = rv[k]; mi = ri[k]; }
            bfar = mi; bc[0] = xs[mi]; bc[1] = ys[mi]; bc[2] = zs[mi];
        }
        __syncthreads();
    }
}

// ---------------------------------------------------------------------------
// K2: ball query (first NSAMPLE indices with d^2<=r^2, ascending) + grouping.
// One wave per query point; ballot/popc stream compaction with early exit.
// Builds X1 rows [M x 68]: (xyz - center, 64 point features, zero pad).
// ---------------------------------------------------------------------------
__global__ __launch_bounds__(256) void group_kernel(const float* __restrict__ xyz,
                                                    const float* __restrict__ points,
                                                    const float* __restrict__ newxyz,
                                                    float* __restrict__ X1) {
    __shared__ int sel[8][NSAMPLE];
    const int tid  = threadIdx.x;
    const int w    = tid >> 5;
    const int lane = tid & 31;
    const int qid  = blockIdx.x * 8 + w;
    const int b    = qid >> 10;
    const float* xb = xyz + (size_t)b * NPTS * 3;

    const float qx = newxyz[(size_t)qid * 3 + 0];
    const float qy = newxyz[(size_t)qid * 3 + 1];
    const float qz = newxyz[(size_t)qid * 3 + 2];
    const float r2 = 0.2f * 0.2f;

    int cnt = 0;
    for (int basei = 0; basei < NPTS && cnt < NSAMPLE; basei += 32) {
        int i = basei + lane;
        float dx = xb[i * 3 + 0] - qx;
        float dy = xb[i * 3 + 1] - qy;
        float dz = xb[i * 3 + 2] - qz;
        bool hit = (dx * dx + dy * dy + dz * dz) <= r2;
        unsigned mask = (unsigned)__ballot(hit);
        int pos = cnt + __popc(mask & ((1u << lane) - 1u));
        if (hit && pos < NSAMPLE) sel[w][pos] = i;
        cnt += __popc(mask);
    }
    if (cnt == 0 && lane == 0) sel[w][0] = 0;   // degenerate (cannot happen: center is a cloud point)
    __syncthreads();

    int nsel = cnt < NSAMPLE ? cnt : NSAMPLE;
    if (nsel == 0) nsel = 1;
    int idx = (lane < nsel) ? sel[w][lane] : sel[w][0];

    size_t m = (size_t)qid * NSAMPLE + lane;
    float* row = X1 + m * LDX1;
    row[0] = xb[idx * 3 + 0] - qx;
    row[1] = xb[idx * 3 + 1] - qy;
    row[2] = xb[idx * 3 + 2] - qz;
    const float* p = points + ((size_t)b * NPTS + idx) * 64;
#pragma unroll 8
    for (int c = 0; c < 64; ++c) row[3 + c] = p[c];
    row[67] = 0.f;
}

// ---------------------------------------------------------------------------
// K3: GEMM Y[M,N] = act(X) @ W^T + bias using V_WMMA_F32_16X16X4_F32, with
// optional fused input BN(scale/shift)+ReLU applied while staging X into LDS,
// and fused per-channel sum / sum-of-squares accumulation for the next BN.
// Block: 256 threads = 8 waves; each wave -> 16 rows x 64 cols (4 acc tiles).
// W is staged into LDS as interleaved k-pairs (v2f) so each WMMA B fragment
// is a single aligned ds_load_b64; A fragments are b64 loads from Xs.
// KCP (padded K, mult of 4) and NTOT (output channels) are compile-time.
// ---------------------------------------------------------------------------
template <int KCP, int NTOT>
__global__ __launch_bounds__(256) void conv_bn_gemm_kernel(
    const float* __restrict__ X,
    const float* __restrict__ in_scale, const float* __restrict__ in_shift, int apply_norm,
    const float* __restrict__ W, int kw,
    const float* __restrict__ bias,
    float* __restrict__ Y, float* __restrict__ ssum, float* __restrict__ ssq) {
    constexpr int KH = KCP / 2;                 // # of interleaved k-pairs
    __shared__ __align__(16) float Xs[128 * 68];   // stride 68 -> conflict-free A reads
    __shared__ __align__(16) v2f   Wv[KH * 64];    // [k/2][n_local] interleaved pairs
    __shared__ float st_s[64];
    __shared__ float st_q[64];

    const int tid  = threadIdx.x;
    const int m0   = blockIdx.x * 128;
    const int yoff = blockIdx.y * 64;

    if (tid < 64) { st_s[tid] = 0.f; st_q[tid] = 0.f; }

    // stage W^T tile as interleaved k-pairs (zero-pad k >= kw)
    for (int t = tid; t < 64 * KH; t += 256) {
        int n = t / KH, kp = t - n * KH;
        int k = kp * 2;
        v2f wv;
        wv.x = (k     < kw) ? W[(size_t)(yoff + n) * kw + k    ] : 0.f;
        wv.y = (k + 1 < kw) ? W[(size_t)(yoff + n) * kw + k + 1] : 0.f;
        Wv[kp * 64 + n] = wv;
    }
    // stage X tile with optional BN+ReLU
    for (int t = tid; t < 128 * KCP; t += 256) {
        int row = t / KCP, col = t - row * KCP;
        float v = X[(size_t)(m0 + row) * KCP + col];
        if (apply_norm) v = fmaxf(fmaf(v, in_scale[col], in_shift[col]), 0.f);
        Xs[row * 68 + col] = v;
    }
    __syncthreads();

    const int lane = tid & 31;
    const int w    = tid >> 5;
    const int r    = lane & 15;
    const int koff = (lane >> 4) << 1;        // lanes 0-15: K+0/K+1, lanes 16-31: K+2/K+3
    const int mrow = w * 16;

    v8f acc[4] = {};
#pragma unroll
    for (int kk = 0; kk < KCP; kk += 4) {
        v2f a = *(const v2f*)&Xs[(mrow + r) * 68 + kk + koff];
        const v2f* wb = &Wv[((kk + koff) >> 1) * 64 + r];
#pragma unroll
        for (int g = 0; g < 4; ++g) {
            acc[g] = __builtin_amdgcn_wmma_f32_16x16x4_f32(
                false, a, false, wb[g * 16], (short)0, acc[g], false, false);
        }
    }

    const int rowadd = (lane < 16) ? 0 : 8;
#pragma unroll
    for (int g = 0; g < 4; ++g) {
        int nl = g * 16 + r;
        int ng = yoff + nl;
        float bz = bias[ng];
        float s = 0.f, q = 0.f;
#pragma unroll
        for (int i = 0; i < 8; ++i) {
            float v = acc[g][i] + bz;
            int row = m0 + mrow + i + rowadd;
            Y[(size_t)row * NTOT + ng] = v;
            s += v; q += v * v;
        }
        atomicAdd(&st_s[nl], s);
        atomicAdd(&st_q[nl], q);
    }
    __syncthreads();
    if (tid < 64) {
        atomicAdd(&ssum[yoff + tid], st_s[tid]);
        atomicAdd(&ssq[yoff + tid], st_q[tid]);
    }
}

// ---------------------------------------------------------------------------
// K4: per-channel BN stats -> (scale, shift)
// ---------------------------------------------------------------------------
__global__ void bnstats_kernel(const float* __restrict__ sum, const float* __restrict__ sq,
                               const float* __restrict__ g, const float* __restrict__ beta,
                               float* __restrict__ scale, float* __restrict__ shift,
                               int n, float invM) {
    int i = blockIdx.x * blockDim.x + threadIdx.x;
    if (i < n) {
        float mean = sum[i] * invM;
        float var  = sq[i] * invM - mean * mean;
        float sc   = g[i] * rsqrtf(var + 1e-5f);
        scale[i] = sc;
        shift[i] = beta[i] - mean * sc;
    }
}

// ---------------------------------------------------------------------------
// K5: final BN+ReLU fused with max over the NSAMPLE axis -> new_points[b,s,o]
// ---------------------------------------------------------------------------
__global__ __launch_bounds__(256) void maxpool_kernel(const float* __restrict__ Y3,
                                                      const float* __restrict__ scale,
                                                      const float* __restrict__ shift,
                                                      float* __restrict__ out) {
    int t  = blockIdx.x * 256 + threadIdx.x;     // over NQ*128
    int o  = t & 127;
    int bs = t >> 7;
    const float* yb = Y3 + (size_t)bs * NSAMPLE * 128 + o;
    float sc = scale[o], sh = shift[o];
    float m = -3.0e38f;
#pragma unroll 8
    for (int k = 0; k < NSAMPLE; ++k)
        m = fmaxf(m, fmaf(yb[k * 128], sc, sh));
    out[t] = fmaxf(m, 0.f);
}

// ---------------------------------------------------------------------------
// Launcher
// ---------------------------------------------------------------------------
extern "C" void kernel_launch(void* const* d_in, const int* in_sizes, int n_in,
                              void* d_out, int out_size, void* d_ws, size_t ws_size,
                              hipStream_t stream) {
    const float* xyz    = (const float*)d_in[0];
    const float* points = (const float*)d_in[1];
    const float* w1     = (const float*)d_in[2];
    const float* b1     = (const float*)d_in[3];
    const float* g1     = (const float*)d_in[4];
    const float* beta1  = (const float*)d_in[5];
    const float* w2     = (const float*)d_in[6];
    const float* b2     = (const float*)d_in[7];
    const float* g2     = (const float*)d_in[8];
    const float* beta2  = (const float*)d_in[9];
    const float* w3     = (const float*)d_in[10];
    const float* b3     = (const float*)d_in[11];
    const float* g3     = (const float*)d_in[12];
    const float* beta3  = (const float*)d_in[13];

    float* out      = (float*)d_out;
    float* newxyz   = out;                 // 16*1024*3
    float* newpts   = out + NEWXYZ_N;      // 16*1024*128

    char* ws = (char*)d_ws;
    // workspace layout
    const size_t OFF_FPS   = 0;                                    // 16K ints
    const size_t OFF_STATS = 65536;                                // 512 floats
    const size_t OFF_SS    = 67584;                                // 512 floats
    const size_t OFF_X1    = 131072;                               // M x 68 f32
    const size_t OFF_Y     = OFF_X1 + MROWS * LDX1 * sizeof(float);

    int*   fps_idx = (int*)(ws + OFF_FPS);
    float* stb     = (float*)(ws + OFF_STATS);
    float* sum1 = stb +   0, *sq1 = stb +  64;
    float* sum2 = stb + 128, *sq2 = stb + 192;
    float* sum3 = stb + 256, *sq3 = stb + 384;   // 128 ch each
    float* ssb     = (float*)(ws + OFF_SS);
    float* scale1 = ssb +   0, *shift1 = ssb +  64;
    float* scale2 = ssb + 128, *shift2 = ssb + 192;
    float* scale3 = ssb + 256, *shift3 = ssb + 384;

    float* X1 = (float*)(ws + OFF_X1);   // M x 68
    float* Y1 = (float*)(ws + OFF_Y);    // M x 64
    float* Y2 = X1;                      // reuse X1 region (M x 64)
    float* Y3 = Y1;                      // reuse Y region  (M x 128)

    const float invM = 1.0f / (float)MROWS;

    zero_kernel<<<2, 256, 0, stream>>>(stb, 512);

    fps_kernel<<<NBATCH, 512, 0, stream>>>(xyz, newxyz, fps_idx);

    group_kernel<<<NQ / 8, 256, 0, stream>>>(xyz, points, newxyz, X1);

    // conv1: X1[M,68] @ w1^T(67->pad68) -> Y1[M,64]; stats1
    conv_bn_gemm_kernel<68, 64><<<dim3(MROWS / 128, 1), 256, 0, stream>>>(
        X1, nullptr, nullptr, 0, w1, 67, b1, Y1, sum1, sq1);
    bnstats_kernel<<<1, 64, 0, stream>>>(sum1, sq1, g1, beta1, scale1, shift1, 64, invM);

    // conv2: relu(bn(Y1)) @ w2^T -> Y2[M,64]; stats2
    conv_bn_gemm_kernel<64, 64><<<dim3(MROWS / 128, 1), 256, 0, stream>>>(
        Y1, scale1, shift1, 1, w2, 64, b2, Y2, sum2, sq2);
    bnstats_kernel<<<1, 64, 0, stream>>>(sum2, sq2, g2, beta2, scale2, shift2, 64, invM);

    // conv3: relu(bn(Y2)) @ w3^T -> Y3[M,128] (two 64-col halves); stats3
    conv_bn_gemm_kernel<64, 128><<<dim3(MROWS / 128, 2), 256, 0, stream>>>(
        Y2, scale2, shift2, 1, w3, 64, b3, Y3, sum3, sq3);
    bnstats_kernel<<<1, 128, 0, stream>>>(sum3, sq3, g3, beta3, scale3, shift3, 128, invM);

    // final bn+relu+maxpool over samples
    maxpool_kernel<<<(NQ * 128) / 256, 256, 0, stream>>>(Y3, scale3, shift3, newpts);
}